// ProfessionalEQ_40192303956726
// MI455X (gfx1250) — compile-verified
//
#include <hip/hip_runtime.h>

// ---------------------------------------------------------------------------
// Biquad peaking EQ over (32,2,441000) fp32, MI455X (gfx1250, wave32).
//
// Blocked parallel scan of the order-2 IIR:
//   f[n] = b0 x[n] + b1 x[n-1] + b2 x[n-2]            (parallel FIR forcing)
//   y[n] = f[n] - a1 y[n-1] - a2 y[n-2]               (recurrence)
// Zero-state response of a 16-sample tile is a lower-triangular 16x16 matmul
// W[i][j] = w[i-j] (w = impulse response of 1/(1+a1 z^-1+a2 z^-2)), computed
// with V_WMMA_F32_16X16X4_F32 (4 k-slices x 16 tiles = 256 samples / group).
// Cross-tile / cross-chunk state propagation is an affine 2x2 recurrence with
// a CONSTANT matrix (entries = w[k]), solved by a per-sequence chunk scan
// between two WMMA passes.
//
// Memory: read x twice (2nd read hits 192MB L2) + write y once (~340MB total
// => ~15us at 23.3 TB/s).  Interior chunks (214/216) take a branch-free path:
// 8 aligned global_load_b64 per lane per 256-sample group with constant
// immediate offsets; tile-state extraction uses v_readlane (uniform lane).
// ---------------------------------------------------------------------------

typedef __attribute__((ext_vector_type(2))) float v2f;
typedef __attribute__((ext_vector_type(4))) float v4f;
typedef __attribute__((ext_vector_type(8))) float v8f;

#define T_LEN 441000
#define NSEQ  64
#define CH    2048                      // samples per chunk (one wave)
#define GRP   256                       // samples per WMMA group (16 tiles x 16)
#define NGRP  (CH / GRP)                // 8
#define NCH   ((T_LEN + CH - 1) / CH)   // 216 chunks per sequence

// workspace (float) layout
#define WS_WL     5                     // [5..7] w[CH], w[CH-1], w[CH-2]
#define WS_W      8                     // [8..24] w[0..16]
#define WS_CHUNKD 32                    // NSEQ*NCH*2 : zero-entry chunk exit states
#define WS_ENTRY  (32 + NSEQ * NCH * 2) // NSEQ*NCH*2 : scanned chunk entry states

__device__ __forceinline__ float lane_read(float v, int srcLane) {
  return __builtin_bit_cast(float,
           __builtin_amdgcn_readlane(__builtin_bit_cast(int, v), srcLane));
}

// ---------------------------------------------------------------------------
__global__ void eq_setup(const int* pf, const int* pg, const int* pq, float* ws) {
  if (threadIdx.x != 0 || blockIdx.x != 0) return;

  float fr = fminf(fmaxf((float)pf[0], 20.f), 20000.f);
  float lmn = __logf(20.f), lmx = __logf(20000.f);
  fr = (__logf(fmaxf(fr, 1e-6f)) - lmn) / (lmx - lmn) * (20000.f - 20.f) + 20.f;
  float gn = fminf(fmaxf((float)pg[0], -30.f), 30.f);
  float q = fminf(fmaxf((float)pq[0], 0.1f), 30.f);
  float qmn = __logf(0.1f), qmx = __logf(30.f);
  q = (__logf(fmaxf(q, 1e-6f)) - qmn) / (qmx - qmn) * (30.f - 0.1f) + 0.1f;

  float w0 = 2.f * 3.14159265358979f * fr / 44100.f;
  float A  = __powf(10.f, gn / 40.f);
  float al = __sinf(w0) / (2.f * q);
  float cs = __cosf(w0);
  float a0 = 1.f + al / A;
  float b0 = (1.f + al * A) / a0;
  float b1 = (-2.f * cs) / a0;
  float b2 = (1.f - al * A) / a0;
  float a1 = b1;
  float a2 = (1.f - al / A) / a0;
  ws[0] = b0; ws[1] = b1; ws[2] = b2; ws[3] = a1; ws[4] = a2;

  float wm1 = 1.f, wm2 = 0.f;           // w[0]=1, w[n] = -a1 w[n-1] - a2 w[n-2]
  ws[WS_W + 0] = 1.f;
  for (int i = 1; i <= CH; ++i) {
    float wi = -a1 * wm1 - a2 * wm2;
    if (i <= 16)    ws[WS_W + i] = wi;
    if (i == CH-2)  ws[WS_WL + 2] = wi;
    if (i == CH-1)  ws[WS_WL + 1] = wi;
    if (i == CH)    ws[WS_WL + 0] = wi;
    wm2 = wm1; wm1 = wi;
  }
}

// ---------------------------------------------------------------------------
// PASS=0: zero-entry chunk exit states.  PASS=1: entry states -> output.
// One wave per 2048-sample chunk; 8 WMMA groups of 256 samples.
// ---------------------------------------------------------------------------
template <int PASS>
__global__ __launch_bounds__(256) void eq_pass(const float* __restrict__ x,
                                               float* __restrict__ out,
                                               float* __restrict__ ws) {
  const int lane = threadIdx.x & 31;
  const int wave = blockIdx.x * 8 + (threadIdx.x >> 5);
  const int seq   = wave / NCH;
  const int chunk = wave - seq * NCH;

  const float* xs = x   + (size_t)seq * T_LEN;
  float*       ys = out + (size_t)seq * T_LEN;

  const float b0 = ws[0], b1 = ws[1], b2 = ws[2], a2 = ws[4];
  const float* w = ws + WS_W;                 // w[0..16]
  const float w16 = w[16], w15 = w[15], w14 = w[14];

  const int  col    = lane & 15;              // tile column / A row
  const bool hi     = lane >= 16;
  const int  mofs   = hi ? 8 : 0;
  const int  k0base = hi ? 2 : 0;

  // A fragments: W[row][k], f32 16x4 layout (lanes 0-15: k0,k1; 16-31: k2,k3)
  v2f afrag[4];
#pragma unroll
  for (int kt = 0; kt < 4; ++kt) {
    const int k0 = kt * 4 + k0base;
    afrag[kt].x = (col >= k0)     ? w[col - k0]     : 0.f;
    afrag[kt].y = (col >= k0 + 1) ? w[col - k0 - 1] : 0.f;
  }

  // per-lane homogeneous correction coefficients (pass 2 only)
  float pcf[8], qcf[8];
  if (PASS == 1) {
#pragma unroll
    for (int r = 0; r < 8; ++r) {
      const int m = r + mofs;
      pcf[r] = w[m + 1];                      // p[m]
      qcf[r] = w[m];                          // q[m]/-a2
    }
  }

  float y1, y2;                               // running (y[n-1], y[n-2]), uniform
  if (PASS == 0) { y1 = 0.f; y2 = 0.f; }
  else { y1 = ws[WS_ENTRY + wave * 2 + 0]; y2 = ws[WS_ENTRY + wave * 2 + 1]; }

  const long base = (long)chunk * CH;
  // interior chunks never touch x[<0] or x[>=T_LEN]
  const bool fast = (base >= 2) && (base + CH <= T_LEN);
  // lane base pointer: x at (base + col*16 + k0base - 2); 8-byte aligned.
  const float* p = xs + base + col * 16 + k0base - 2;

  // shared per-group tail: 16-tile scan via readlane + (pass2) output store
  auto tail = [&](const v8f& c, int g, bool safeStore) {
    float ty1 = 0.f, ty2 = 0.f;
#pragma unroll
    for (int t = 0; t < 16; ++t) {
      // rows 15/14 of tile t live in c[7]/c[6] on lane 16+t (uniform lane)
      const float v15 = lane_read(c[7], 16 + t);
      const float v14 = lane_read(c[6], 16 + t);
      if (PASS == 1 && col == t) { ty1 = y1; ty2 = y2; }
      const float n1 = v15 + w16 * y1 - a2 * w15 * y2;
      const float n2 = v14 + w15 * y1 - a2 * w14 * y2;
      y1 = n1; y2 = n2;
    }
    if (PASS == 1) {
      float yv[8];
#pragma unroll
      for (int r = 0; r < 8; ++r)
        yv[r] = c[r] + pcf[r] * ty1 - a2 * qcf[r] * ty2;
      const long sb = base + (long)g * GRP + (long)col * 16 + mofs;
      if (safeStore || sb + 8 <= T_LEN) {
        v4f lo4 = { yv[0], yv[1], yv[2], yv[3] };
        v4f hi4 = { yv[4], yv[5], yv[6], yv[7] };
        *(v4f*)(ys + sb)     = lo4;
        *(v4f*)(ys + sb + 4) = hi4;
      } else {
#pragma unroll
        for (int r = 0; r < 8; ++r)
          if (sb + r < T_LEN) ys[sb + r] = yv[r];
      }
    }
  };

  if (fast) {
#pragma unroll
    for (int g = 0; g < NGRP; ++g) {
      v8f c = {};
#pragma unroll
      for (int kt = 0; kt < 4; ++kt) {
        // constant offsets -> global_load_b64 with immediate IOFFSET
        v2f xa = *(const v2f*)(p + g * GRP + kt * 4);      // x[s0-2], x[s0-1]
        v2f xb = *(const v2f*)(p + g * GRP + kt * 4 + 2);  // x[s0],   x[s0+1]
        v2f bfrag;
        bfrag.x = b0 * xb.x + b1 * xa.y + b2 * xa.x;
        bfrag.y = b0 * xb.y + b1 * xb.x + b2 * xa.y;
        c = __builtin_amdgcn_wmma_f32_16x16x4_f32(
                false, afrag[kt], false, bfrag, (short)0, c, false, false);
      }
      tail(c, g, true);
    }
  } else {
    // boundary chunks (first / last of each sequence): rare, keep compact
#pragma unroll 1
    for (int g = 0; g < NGRP; ++g) {
      const long gb = base + (long)g * GRP;
      v8f c = {};
#pragma unroll
      for (int kt = 0; kt < 4; ++kt) {
        const int  k0 = kt * 4 + k0base;
        const long s0 = gb + (long)col * 16 + k0;
        float xm2 = (s0 - 2 >= 0 && s0 - 2 < T_LEN) ? xs[s0 - 2] : 0.f;
        float xm1 = (s0 - 1 >= 0 && s0 - 1 < T_LEN) ? xs[s0 - 1] : 0.f;
        float x0  = (s0     < T_LEN)                ? xs[s0]     : 0.f;
        float x1v = (s0 + 1 < T_LEN)                ? xs[s0 + 1] : 0.f;
        v2f bfrag;
        bfrag.x = b0 * x0  + b1 * xm1 + b2 * xm2;
        bfrag.y = b0 * x1v + b1 * x0  + b2 * xm1;
        c = __builtin_amdgcn_wmma_f32_16x16x4_f32(
                false, afrag[kt], false, bfrag, (short)0, c, false, false);
      }
      tail(c, g, false);
    }
  }

  if (PASS == 0 && lane == 0) {
    ws[WS_CHUNKD + wave * 2 + 0] = y1;
    ws[WS_CHUNKD + wave * 2 + 1] = y2;
  }
}

// ---------------------------------------------------------------------------
// Per-sequence scan over chunk states.
// M_CH = [[w[CH], -a2 w[CH-1]], [w[CH-1], -a2 w[CH-2]]] (constant).
// ---------------------------------------------------------------------------
__global__ void eq_scan(float* __restrict__ ws) {
  const int seq = blockIdx.x * blockDim.x + threadIdx.x;
  if (seq >= NSEQ) return;
  const float a2  = ws[4];
  const float wl0 = ws[WS_WL + 0], wl1 = ws[WS_WL + 1], wl2 = ws[WS_WL + 2];
  float y1 = 0.f, y2 = 0.f;
  for (int c = 0; c < NCH; ++c) {
    const int i = seq * NCH + c;
    ws[WS_ENTRY + i * 2 + 0] = y1;
    ws[WS_ENTRY + i * 2 + 1] = y2;
    const float d1 = ws[WS_CHUNKD + i * 2 + 0];
    const float d2 = ws[WS_CHUNKD + i * 2 + 1];
    const float n1 = d1 + wl0 * y1 - a2 * wl1 * y2;
    const float n2 = d2 + wl1 * y1 - a2 * wl2 * y2;
    y1 = n1; y2 = n2;
  }
}

// ---------------------------------------------------------------------------
extern "C" void kernel_launch(void* const* d_in, const int* in_sizes, int n_in,
                              void* d_out, int out_size, void* d_ws, size_t ws_size,
                              hipStream_t stream) {
  const float* x  = (const float*)d_in[0];
  const int*   pf = (const int*)d_in[1];
  const int*   pg = (const int*)d_in[2];
  const int*   pq = (const int*)d_in[3];
  float* out = (float*)d_out;
  float* ws  = (float*)d_ws;

  eq_setup<<<1, 32, 0, stream>>>(pf, pg, pq, ws);

  const int totalWaves = NSEQ * NCH;                 // 13824 chunk-waves
  const int blocks = totalWaves / 8;                 // 8 waves / 256-thread block

  eq_pass<0><<<blocks, 256, 0, stream>>>(x, out, ws);
  eq_scan<<<2, 32, 0, stream>>>(ws);
  eq_pass<1><<<blocks, 256, 0, stream>>>(x, out, ws);
}